// VQVAE_10892037063020
// MI455X (gfx1250) — compile-verified
//
#include <hip/hip_runtime.h>
#include <hip/hip_bf16.h>

typedef float v2f __attribute__((ext_vector_type(2)));
typedef float v8f __attribute__((ext_vector_type(8)));

#define B_SZ  16
#define C_IN  256
#define T_SZ  8192
#define D_SZ  64
#define K_SZ  512

#define WAVES_PER_WG 8
#define T_PER_WAVE   64                            // 4 t-tiles of 16 per wave
#define NTT          4
#define T_PER_WG     (WAVES_PER_WG * T_PER_WAVE)   // 512
#define ZPAD         68                            // [t][d] row stride (floats)

// ---------------------------------------------------------------------------
// Kernel 0: cc[k] = ||codebook[k]||^2   (512 values into d_ws)
// ---------------------------------------------------------------------------
__global__ void vq_cc_kernel(const float* __restrict__ cb, float* __restrict__ cc) {
    int k = blockIdx.x * blockDim.x + threadIdx.x;
    if (k >= K_SZ) return;
    const float4* row = (const float4*)(cb + (size_t)k * D_SZ);
    float s = 0.f;
    #pragma unroll
    for (int i = 0; i < D_SZ / 4; ++i) {
        float4 v = row[i];
        s += v.x * v.x + v.y * v.y + v.z * v.z + v.w * v.w;
    }
    cc[k] = s;
}

// ---------------------------------------------------------------------------
// Main fused kernel: z = W*x + b ; out[b,t,{0,1}] = min_k ||z - e_k||^2
// One wave owns 64 t-columns of one batch. All GEMM work runs on
// v_wmma_f32_16x16x4_f32 (exact fp32 matrix path on CDNA5); z is exchanged
// through a per-wave transposed LDS tile ([t][d], pad 68) so the GEMM2
// B-operand is a single aligned ds_load_b64 per tile.
// ---------------------------------------------------------------------------
__global__ __launch_bounds__(256, 1) void vq_main_kernel(
    const float* __restrict__ x,    // [B, C_IN, T]
    const float* __restrict__ W,    // [D, C_IN]
    const float* __restrict__ bias, // [D]
    const float* __restrict__ cb,   // [K, D]
    const float* __restrict__ cc,   // [K]  (precomputed ||e_k||^2)
    float* __restrict__ out)        // [B, T, 2]
{
    __shared__ float zlds[WAVES_PER_WG * T_PER_WAVE * ZPAD];   // ~136 KB

    const int wave = threadIdx.x >> 5;
    const int lane = threadIdx.x & 31;
    const int n    = lane & 15;
    const int hi   = lane >> 4;      // which half of the wave (row/k split)

    const int b  = blockIdx.x >> 4;                               // 16 batches
    const int t0 = (blockIdx.x & 15) * T_PER_WG + wave * T_PER_WAVE;

    float* zw = zlds + wave * (T_PER_WAVE * ZPAD);

    // ---------------- GEMM1: z(64x64) = W(64x256) * x_tile(256x64) ----------
    v8f acc[4][NTT];
    #pragma unroll
    for (int dt = 0; dt < 4; ++dt)
        #pragma unroll
        for (int tt = 0; tt < NTT; ++tt)
            acc[dt][tt] = (v8f){};

    const float* xb = x + (size_t)b * C_IN * T_SZ;

    #pragma unroll 2
    for (int c0 = 0; c0 < C_IN; c0 += 4) {
        // B operand: x[c0 + 2*hi + v][t]  (shared across the 4 d-tiles)
        v2f bm[NTT];
        #pragma unroll
        for (int tt = 0; tt < NTT; ++tt) {
            const int tcol = t0 + tt * 16 + n;
            bm[tt].x = xb[(size_t)(c0 + 2 * hi + 0) * T_SZ + tcol];
            bm[tt].y = xb[(size_t)(c0 + 2 * hi + 1) * T_SZ + tcol];
        }
        // prefetch the x rows needed two k-steps ahead (global_prefetch_b8)
        if (c0 + 8 < C_IN)
            __builtin_prefetch(&xb[(size_t)(c0 + 8 + 2 * hi) * T_SZ + t0 + n], 0, 3);
        #pragma unroll
        for (int dt = 0; dt < 4; ++dt) {
            // A operand: W[dt*16 + m][c0 + 2*hi + {0,1}] -> one b64 load
            v2f a = *(const v2f*)(W + (size_t)(dt * 16 + n) * C_IN + c0 + 2 * hi);
            #pragma unroll
            for (int tt = 0; tt < NTT; ++tt)
                acc[dt][tt] = __builtin_amdgcn_wmma_f32_16x16x4_f32(
                    false, a, false, bm[tt], (short)0, acc[dt][tt], false, false);
        }
    }

    // Epilogue: add bias, accumulate ||z||^2 column partials, stash z to LDS
    // transposed as [t][d] so v=0..7 are 8 consecutive floats in a t-row.
    float zz[NTT] = {0.f, 0.f, 0.f, 0.f};
    #pragma unroll
    for (int dt = 0; dt < 4; ++dt) {
        float bv[8];
        #pragma unroll
        for (int v = 0; v < 8; ++v)
            bv[v] = bias[dt * 16 + 8 * hi + v];
        #pragma unroll
        for (int tt = 0; tt < NTT; ++tt) {
            float zv[8];
            #pragma unroll
            for (int v = 0; v < 8; ++v) {
                zv[v] = acc[dt][tt][v] + bv[v];
                zz[tt] += zv[v] * zv[v];
            }
            float* dst = &zw[(tt * 16 + n) * ZPAD + dt * 16 + 8 * hi];
            *(float4*)(dst + 0) = make_float4(zv[0], zv[1], zv[2], zv[3]);
            *(float4*)(dst + 4) = make_float4(zv[4], zv[5], zv[6], zv[7]);
        }
    }

    // ---------------- GEMM2 + running min over the 512 codes ----------------
    float rm[NTT] = {3.0e38f, 3.0e38f, 3.0e38f, 3.0e38f};

    for (int kt = 0; kt < K_SZ / 16; ++kt) {
        v8f s[NTT];
        #pragma unroll
        for (int tt = 0; tt < NTT; ++tt) s[tt] = (v8f){};
        #pragma unroll 4
        for (int d0 = 0; d0 < D_SZ; d0 += 4) {
            // A: codebook[kt*16 + m][d0 + 2*hi + {0,1}] -> b64 load
            v2f a = *(const v2f*)(cb + (size_t)(kt * 16 + n) * D_SZ + d0 + 2 * hi);
            #pragma unroll
            for (int tt = 0; tt < NTT; ++tt) {
                // B: z[d0 + 2*hi + {0,1}][t] -> one aligned ds_load_b64
                v2f bz = *(const v2f*)&zw[(tt * 16 + n) * ZPAD + d0 + 2 * hi];
                s[tt] = __builtin_amdgcn_wmma_f32_16x16x4_f32(
                    false, a, false, bz, (short)0, s[tt], false, false);
            }
        }
        // d2(k,t) = ||z||^2 - 2*s + cc[k]; fold (cc - 2 s) into running min
        #pragma unroll
        for (int v = 0; v < 8; ++v) {
            const float ccv = cc[kt * 16 + v + 8 * hi];
            #pragma unroll
            for (int tt = 0; tt < NTT; ++tt)
                rm[tt] = fminf(rm[tt], __builtin_fmaf(-2.f, s[tt][v], ccv));
        }
    }

    // Combine the two lane-halves: rows (k subsets / d subsets) are split
    // between lane n and lane n+16 for the same column t.
    #pragma unroll
    for (int tt = 0; tt < NTT; ++tt) {
        rm[tt] = fminf(rm[tt], __shfl_xor(rm[tt], 16, 32));
        zz[tt] += __shfl_xor(zz[tt], 16, 32);
    }

    // out[b,t,0] == out[b,t,1] == min_k ||z - e_k||^2  (forward values equal)
    if (hi == 0) {
        float2* o = (float2*)out;
        #pragma unroll
        for (int tt = 0; tt < NTT; ++tt) {
            const int t = t0 + tt * 16 + n;
            float val = fmaxf(zz[tt] + rm[tt], 0.f);
            o[(size_t)b * T_SZ + t] = make_float2(val, val);
        }
    }
}

// ---------------------------------------------------------------------------
extern "C" void kernel_launch(void* const* d_in, const int* in_sizes, int n_in,
                              void* d_out, int out_size, void* d_ws, size_t ws_size,
                              hipStream_t stream) {
    const float* x    = (const float*)d_in[0];  // [16, 256, 8192]
    const float* W    = (const float*)d_in[1];  // [64, 256]
    const float* bias = (const float*)d_in[2];  // [64]
    const float* cb   = (const float*)d_in[3];  // [512, 64]
    float* cc = (float*)d_ws;                   // 512 floats scratch

    vq_cc_kernel<<<2, 256, 0, stream>>>(cb, cc);

    const int grid = B_SZ * (T_SZ / T_PER_WG);  // 16 * 16 = 256 workgroups
    vq_main_kernel<<<grid, 256, 0, stream>>>(x, W, bias, cb, cc, (float*)d_out);
}